// EdgeConv2d_global_80075370266798
// MI455X (gfx1250) — compile-verified
//
#include <hip/hip_runtime.h>

// EdgeConv restructured:
//   out[b,o,n] = relu( max_k  (W1-W2)·x[:,i(n,k)] + W2·x[:,j(n,k)] + (W3·mean(x)+bias) )
// Phase 1: tiny prep (WcatT = [[W1-W2];[W2]]^T, per-channel mean, c0 vector)
// Phase 2: WMMA f32 GEMM  Yt[b][n][0:128] = Wcat(128x64) @ x(64xN), stored node-major
// Phase 3: edge gather + add + max + relu  (L2-resident gathers, 256B coalesced)

#define B_    2
#define C_    64
#define N_    16384
#define K_    16
#define COUT_ 64
#define CCAT_ 128   // rows 0..63 = (W1-W2)x , rows 64..127 = W2 x

typedef __attribute__((ext_vector_type(2))) float v2f;
typedef __attribute__((ext_vector_type(8))) float v8f;

// ---------------- Phase 1 kernels ----------------

// WcatT[c][m] : transposed concatenated weight, 64 x 128
__global__ void __launch_bounds__(256)
prep_wcat_kernel(const float* __restrict__ W, float* __restrict__ WcatT) {
    int t = blockIdx.x * blockDim.x + threadIdx.x;     // 0 .. 8191
    if (t >= C_ * CCAT_) return;
    int c = t / CCAT_;
    int m = t % CCAT_;
    float v;
    if (m < COUT_) v = W[m * (3 * C_) + c] - W[m * (3 * C_) + C_ + c]; // W1 - W2
    else           v = W[(m - COUT_) * (3 * C_) + C_ + c];             // W2
    WcatT[c * CCAT_ + m] = v;
}

// per-(b,c) mean over N nodes
__global__ void __launch_bounds__(256)
mean_kernel(const float* __restrict__ x, float* __restrict__ gmean) {
    __shared__ float s[256];
    const int bc = blockIdx.x;                  // b*C_ + c
    const float* row = x + (size_t)bc * N_;
    float acc = 0.f;
    for (int n = threadIdx.x; n < N_; n += 256) acc += row[n];
    s[threadIdx.x] = acc;
    __syncthreads();
    for (int off = 128; off > 0; off >>= 1) {
        if (threadIdx.x < off) s[threadIdx.x] += s[threadIdx.x + off];
        __syncthreads();
    }
    if (threadIdx.x == 0) gmean[bc] = s[0] * (1.0f / N_);
}

// c0[b][o] = bias[o] + sum_c W3[o][c] * gmean[b][c]
__global__ void __launch_bounds__(128)
c0_kernel(const float* __restrict__ W, const float* __restrict__ bias,
          const float* __restrict__ gmean, float* __restrict__ c0) {
    int t = threadIdx.x;                        // 0..127
    if (t >= B_ * COUT_) return;
    int b = t / COUT_, o = t % COUT_;
    float acc = bias[o];
    #pragma unroll 8
    for (int c = 0; c < C_; ++c)
        acc += W[o * (3 * C_) + 2 * C_ + c] * gmean[b * C_ + c];
    c0[t] = acc;
}

// ---------------- Phase 2: WMMA f32 GEMM ----------------
// D(16 nodes x 16 out-chans) = A(16 nodes x 4 chans) * B(4 chans x 16 out-chans), K-loop over 64.
// A layout (ISA 32-bit A 16x4): lanes0-15 M=0..15 K={0,1}; lanes16-31 M=0..15 K={2,3}
// B layout (ISA rows striped): lanes0-15 rows {0,1} in VGPR{0,1}; lanes16-31 rows {2,3}
// D layout: VGPR v -> M = v + 8*(lane>=16), N = lane&15
__global__ void __launch_bounds__(128)
gemm_wmma_kernel(const float* __restrict__ x, const float* __restrict__ WcatT,
                 float* __restrict__ Yt) {
    const int lane = threadIdx.x;                       // 0..31 (wave32)
    const int wv   = threadIdx.y;                       // 0..3
    const int n0   = blockIdx.x * 16;                   // node tile base
    const int m0   = (blockIdx.y * 4 + wv) * 16;        // out-channel tile base, [0,128)
    const int b    = blockIdx.z;
    const float* xb = x + (size_t)b * C_ * N_;
    float* yb = Yt + (size_t)b * N_ * CCAT_;
    const int ln = lane & 15;
    const int hi = lane >> 4;                           // half-wave select

    v8f acc = {};
    #pragma unroll
    for (int c = 0; c < C_; c += 4) {
        v2f a, bm;
        // A: M = node (n0+ln), K = c + 2*hi + {0,1}; row-major x => 64B contiguous per half-wave
        a.x = xb[(size_t)(c + 2 * hi    ) * N_ + n0 + ln];
        a.y = xb[(size_t)(c + 2 * hi + 1) * N_ + n0 + ln];
        // B: K rows of WcatT, N = out-channel column (m0+ln)
        bm.x = WcatT[(c + 2 * hi    ) * CCAT_ + m0 + ln];
        bm.y = WcatT[(c + 2 * hi + 1) * CCAT_ + m0 + ln];
        // 8-arg probed pattern: (neg_a, A, neg_b, B, c_mod, C, reuse_a, reuse_b)
        acc = __builtin_amdgcn_wmma_f32_16x16x4_f32(
            false, a, false, bm, (short)0, acc, false, false);
    }

    // Store D tile node-major: Yt[node][chan]; per VGPR two 64B contiguous runs
    #pragma unroll
    for (int v = 0; v < 8; ++v) {
        int node = n0 + v + 8 * hi;
        yb[(size_t)node * CCAT_ + m0 + ln] = acc[v];
    }
}

// ---------------- Phase 3: edge gather + max + relu ----------------
// 64 threads per node (one per out channel); gathers are 256B contiguous per node
// and hit the L2-resident Yt buffer.
__global__ void __launch_bounds__(256)
edge_kernel(const int* __restrict__ eidx, const float* __restrict__ Yt,
            const float* __restrict__ c0, float* __restrict__ out) {
    int t    = blockIdx.x * blockDim.x + threadIdx.x;
    int o    = t & (COUT_ - 1);
    int node = t >> 6;                                  // 0 .. B_*N_-1
    if (node >= B_ * N_) return;
    int b = node / N_;
    int n = node - b * N_;

    const int* e0 = eidx + ((size_t)b * N_ + n) * K_;                        // edge_index[0] -> x_j
    const int* e1 = eidx + (size_t)B_ * N_ * K_ + ((size_t)b * N_ + n) * K_; // edge_index[1] -> x_i
    const float* yb = Yt + (size_t)b * N_ * CCAT_;
    const float bias0 = c0[b * COUT_ + o];

    float m = -3.402823466e38f;
    #pragma unroll
    for (int k = 0; k < K_; ++k) {
        int i = e1[k];   // center  -> y1 = (W1-W2)x
        int j = e0[k];   // neighbor-> y2 = W2 x
        float v = yb[(size_t)i * CCAT_ + o]
                + yb[(size_t)j * CCAT_ + COUT_ + o]
                + bias0;
        m = fmaxf(m, v);
    }
    // max_k relu(v_k) == relu(max_k v_k)  (relu is monotone)
    out[((size_t)b * COUT_ + o) * N_ + n] = fmaxf(m, 0.0f);
}

// ---------------- launch ----------------
extern "C" void kernel_launch(void* const* d_in, const int* in_sizes, int n_in,
                              void* d_out, int out_size, void* d_ws, size_t ws_size,
                              hipStream_t stream) {
    const float* x    = (const float*)d_in[0];   // [B, C, N] f32
    const int*   eidx = (const int*)  d_in[1];   // [2, B, N, K] (int per harness convention)
    const float* W    = (const float*)d_in[2];   // [64, 192] f32
    const float* bias = (const float*)d_in[3];   // [64] f32

    float* ws    = (float*)d_ws;
    float* WcatT = ws;                            //  8192 floats (32 KB)
    float* gmean = ws + 8192;                     //   128 floats
    float* c0    = ws + 8192 + 128;               //   128 floats
    float* Yt    = ws + 8192 + 256;               //  B*N*128 = 4M floats (16 MB)

    prep_wcat_kernel<<<(C_ * CCAT_ + 255) / 256, 256, 0, stream>>>(W, WcatT);
    mean_kernel<<<B_ * C_, 256, 0, stream>>>(x, gmean);
    c0_kernel<<<1, 128, 0, stream>>>(W, bias, gmean, c0);

    dim3 ggrid(N_ / 16, (CCAT_ / 16) / 4, B_);    // 1024 x 2 x 2 blocks
    dim3 gblock(32, 4, 1);                        // 4 waves, one 16x16 out tile each
    gemm_wmma_kernel<<<ggrid, gblock, 0, stream>>>(x, WcatT, Yt);

    edge_kernel<<<(B_ * N_ * COUT_) / 256, 256, 0, stream>>>(eidx, Yt, c0, (float*)d_out);
}